// LearnedLagAttention_26534307955231
// MI455X (gfx1250) — compile-verified
//
#include <hip/hip_runtime.h>
#include <hip/hip_bf16.h>
#include <math.h>

// LearnedLagAttention, MI455X (gfx1250, wave32).
// Shapes fixed by the reference:
//   B=64, LOOKBACK=32, N_STOCKS=256, D_MODEL=512, D_K=128, D_V=128, MAX_LAG=16
//
// Algebraic refactor: fold W_k into q (qk = q@W_k) so scores = qk . emb,
// and fold W_v after the attention sum (context = (sum_s attn*emb) @ W_v^T).
// This removes the two big (B*S)xD_MODELxD_K projections entirely; the kernel
// is then bound by the 67MB lagged-embedding gather (~3us @ 23.3 TB/s), and
// the three remaining tiny f32 GEMMs run on V_WMMA_F32_16X16X4_F32.

#define NB 64
#define LOOKBACK 32
#define NS 256
#define DM 512
#define DK 128
#define DV 128

typedef __attribute__((ext_vector_type(2))) float v2f;
typedef __attribute__((ext_vector_type(8))) float v8f;

// ---------------------------------------------------------------------------
// Kernel 1: per-stock lag params + gates (256 threads, 1 block)
// ---------------------------------------------------------------------------
__global__ __launch_bounds__(NS) void lla_prep(const float* __restrict__ lags,
                                               const float* __restrict__ causal_gates,
                                               float* __restrict__ alpha,
                                               float* __restrict__ gate,
                                               int* __restrict__ lag_f,
                                               int* __restrict__ lag_c) {
    int s = threadIdx.x;
    float lag = 16.0f / (1.0f + expf(-lags[s]));            // sigmoid(lags)*MAX_LAG
    float fl = floorf(lag);
    float ce = ceilf(lag);
    int lfi = (int)fminf(fmaxf(fl, 0.0f), (float)(LOOKBACK - 1));
    int lci = (int)fminf(fmaxf(ce, 0.0f), (float)(LOOKBACK - 1));
    alpha[s] = lag - (float)lfi;
    gate[s]  = 1.0f / (1.0f + expf(-causal_gates[s]));
    lag_f[s] = lfi;
    lag_c[s] = lci;
}

// ---------------------------------------------------------------------------
// WMMA f32 GEMMs: one wave (32 threads) per 16x16 output tile, EXEC all-ones.
// A-operand lane layout (16x4, MxK): lane nl (0..15), half h = lane>>4:
//   v0 = A[nl][k0+2h], v1 = A[nl][k0+2h+1]
// B-operand (4x16, KxN): v0 = B[k0+2h][nl], v1 = B[k0+2h+1][nl]
// D (16x16): VGPR r -> row (r + 8h), col nl.
// ---------------------------------------------------------------------------

// C[M,N] = A[M,K] @ W[N,K]^T   (both row-major; operand loads are symmetric)
__global__ __launch_bounds__(32) void lla_gemm_at(const float* __restrict__ A,
                                                  const float* __restrict__ W,
                                                  float* __restrict__ C,
                                                  int K, int N) {
    const int lane = threadIdx.x;
    const int h  = lane >> 4;
    const int nl = lane & 15;
    const int m0 = blockIdx.x * 16;
    const int n0 = blockIdx.y * 16;

    const float* arow = A + (size_t)(m0 + nl) * K + 2 * h;
    const float* brow = W + (size_t)(n0 + nl) * K + 2 * h;

    v8f acc = {};
#pragma unroll 8
    for (int k0 = 0; k0 < K; k0 += 4) {
        v2f a = *(const v2f*)(arow + k0);
        v2f b = *(const v2f*)(brow + k0);
        acc = __builtin_amdgcn_wmma_f32_16x16x4_f32(false, a, false, b,
                                                    (short)0, acc, false, false);
    }
    float* crow = C + (size_t)(m0 + 8 * h) * N + n0 + nl;
#pragma unroll
    for (int r = 0; r < 8; ++r) crow[(size_t)r * N] = acc[r];
}

// C[M,N] = A[M,K] @ Bm[K,N]    (both row-major)
__global__ __launch_bounds__(32) void lla_gemm_ab(const float* __restrict__ A,
                                                  const float* __restrict__ Bm,
                                                  float* __restrict__ C,
                                                  int K, int N) {
    const int lane = threadIdx.x;
    const int h  = lane >> 4;
    const int nl = lane & 15;
    const int m0 = blockIdx.x * 16;
    const int n0 = blockIdx.y * 16;

    const float* arow = A + (size_t)(m0 + nl) * K + 2 * h;
    const float* bcol = Bm + n0 + nl;

    v8f acc = {};
#pragma unroll 8
    for (int k0 = 0; k0 < K; k0 += 4) {
        v2f a = *(const v2f*)(arow + k0);
        v2f b;
        b.x = bcol[(size_t)(k0 + 2 * h)     * N];
        b.y = bcol[(size_t)(k0 + 2 * h + 1) * N];
        acc = __builtin_amdgcn_wmma_f32_16x16x4_f32(false, a, false, b,
                                                    (short)0, acc, false, false);
    }
    float* crow = C + (size_t)(m0 + 8 * h) * N + n0 + nl;
#pragma unroll
    for (int r = 0; r < 8; ++r) crow[(size_t)r * N] = acc[r];
}

// ---------------------------------------------------------------------------
// Kernel 4: scores[b,s] = (qk[b] . emb[b,s]) * gate[s] / sqrt(DK)
// emb[b,s,d] = lerp(all[b,lf[s],s,d], all[b,lc[s],s,d], alpha[s])
// grid (B, NS/32), 256 threads = 8 waves, 4 stocks per wave, lanes over d.
// ---------------------------------------------------------------------------
__global__ __launch_bounds__(256) void lla_scores(const float* __restrict__ all_embs,
                                                  const float* __restrict__ qk,
                                                  const float* __restrict__ alpha,
                                                  const float* __restrict__ gate,
                                                  const int* __restrict__ lag_f,
                                                  const int* __restrict__ lag_c,
                                                  float* __restrict__ scores) {
    __shared__ float qks[DM];
    const int b = blockIdx.x;
    const int s_base = blockIdx.y * 32;

    for (int i = threadIdx.x; i < DM; i += 256) qks[i] = qk[(size_t)b * DM + i];
    __syncthreads();

    const int wave = threadIdx.x >> 5;
    const int lane = threadIdx.x & 31;

#pragma unroll
    for (int j = 0; j < 4; ++j) {
        const int s = s_base + wave * 4 + j;
        const int lfi = lag_f[s];
        const int lci = lag_c[s];
        const float al = alpha[s];
        const float* ef = all_embs + (((size_t)b * LOOKBACK + lfi) * NS + s) * DM;
        const float* ec = all_embs + (((size_t)b * LOOKBACK + lci) * NS + s) * DM;

        float acc = 0.0f;
#pragma unroll 4
        for (int d = lane; d < DM; d += 32) {
            float vf = ef[d];
            float vc = ec[d];
            acc = fmaf(qks[d], fmaf(al, vc - vf, vf), acc);
        }
        // wave32 reduction
#pragma unroll
        for (int m = 16; m > 0; m >>= 1) acc += __shfl_xor(acc, m, 32);
        if (lane == 0)
            scores[(size_t)b * NS + s] = acc * gate[s] * 0.08838834764831845f; // 1/sqrt(128)
    }
}

// ---------------------------------------------------------------------------
// Kernel 5: softmax over stocks + attention-weighted embedding sum.
// ctx_emb[b,d] = sum_s attn[b,s] * emb[b,s,d]   (re-gather hits L2: 192MB)
// grid B, 256 threads; thread handles d = tid and d + 256.
// ---------------------------------------------------------------------------
__global__ __launch_bounds__(256) void lla_softmax_wsum(const float* __restrict__ all_embs,
                                                        const float* __restrict__ scores,
                                                        const float* __restrict__ alpha,
                                                        const int* __restrict__ lag_f,
                                                        const int* __restrict__ lag_c,
                                                        float* __restrict__ ctx_emb) {
    __shared__ float attn[NS];
    __shared__ float red[NS];
    __shared__ float s_al[NS];
    __shared__ int   s_lf[NS];
    __shared__ int   s_lc[NS];

    const int b = blockIdx.x;
    const int tid = threadIdx.x;

    const float sc = scores[(size_t)b * NS + tid];
    s_al[tid] = alpha[tid];
    s_lf[tid] = lag_f[tid];
    s_lc[tid] = lag_c[tid];

    // max-reduce
    red[tid] = sc;
    __syncthreads();
#pragma unroll
    for (int off = 128; off > 0; off >>= 1) {
        if (tid < off) red[tid] = fmaxf(red[tid], red[tid + off]);
        __syncthreads();
    }
    const float mx = red[0];
    __syncthreads();

    // sum-reduce of exponentials
    const float e = expf(sc - mx);
    red[tid] = e;
    __syncthreads();
#pragma unroll
    for (int off = 128; off > 0; off >>= 1) {
        if (tid < off) red[tid] += red[tid + off];
        __syncthreads();
    }
    attn[tid] = e / red[0];
    __syncthreads();

    // weighted embedding sum (coalesced over d; loops over all 256 stocks)
    float acc0 = 0.0f, acc1 = 0.0f;
    const size_t bbase = (size_t)b * LOOKBACK;
    for (int s = 0; s < NS; ++s) {
        const float a  = attn[s];
        const float al = s_al[s];
        const float* ef = all_embs + ((bbase + s_lf[s]) * NS + s) * DM;
        const float* ec = all_embs + ((bbase + s_lc[s]) * NS + s) * DM;
        float vf0 = ef[tid],       vc0 = ec[tid];
        float vf1 = ef[tid + 256], vc1 = ec[tid + 256];
        acc0 = fmaf(a, fmaf(al, vc0 - vf0, vf0), acc0);
        acc1 = fmaf(a, fmaf(al, vc1 - vf1, vf1), acc1);
    }
    ctx_emb[(size_t)b * DM + tid]       = acc0;
    ctx_emb[(size_t)b * DM + tid + 256] = acc1;
}

// ---------------------------------------------------------------------------
// Launch: prep -> q (WMMA) -> qk (WMMA) -> scores -> softmax+wsum -> out (WMMA)
// ---------------------------------------------------------------------------
extern "C" void kernel_launch(void* const* d_in, const int* in_sizes, int n_in,
                              void* d_out, int out_size, void* d_ws, size_t ws_size,
                              hipStream_t stream) {
    const float* query    = (const float*)d_in[0];  // (64, 512)
    const float* all_embs = (const float*)d_in[1];  // (64, 32, 256, 512)
    const float* W_q      = (const float*)d_in[2];  // (128, 512)
    const float* W_k      = (const float*)d_in[3];  // (128, 512)
    const float* W_v      = (const float*)d_in[4];  // (128, 512)
    const float* lags     = (const float*)d_in[5];  // (256,)
    const float* cgates   = (const float*)d_in[6];  // (256,)
    float* out = (float*)d_out;                     // (64, 128)

    char* ws = (char*)d_ws;
    float* alpha  = (float*)(ws + 0);                 //  1 KB
    float* gate   = (float*)(ws + 1024);              //  1 KB
    int*   lag_f  = (int*)(ws + 2048);                //  1 KB
    int*   lag_c  = (int*)(ws + 3072);                //  1 KB
    float* qbuf   = (float*)(ws + 4096);              // 64*128  =  32 KB
    float* qkbuf  = (float*)(ws + 4096 + 32768);      // 64*512  = 128 KB
    float* scores = (float*)(ws + 4096 + 32768 + 131072);           // 64 KB
    float* ctxemb = (float*)(ws + 4096 + 32768 + 131072 + 65536);   // 128 KB

    // 1) per-stock lag params / gates
    lla_prep<<<1, NS, 0, stream>>>(lags, cgates, alpha, gate, lag_f, lag_c);

    // 2) q = query @ W_q^T  : (64x128, K=512)   [WMMA]
    lla_gemm_at<<<dim3(NB / 16, DK / 16), 32, 0, stream>>>(query, W_q, qbuf, DM, DK);

    // 3) qk = q @ W_k       : (64x512, K=128)   [WMMA]  (folds K-projection)
    lla_gemm_ab<<<dim3(NB / 16, DM / 16), 32, 0, stream>>>(qbuf, W_k, qkbuf, DK, DM);

    // 4) scores[b,s] = gate[s]/sqrt(dk) * (qk[b] . lerped emb[b,s])
    lla_scores<<<dim3(NB, NS / 32), 256, 0, stream>>>(all_embs, qkbuf, alpha, gate,
                                                      lag_f, lag_c, scores);

    // 5) softmax over stocks + attention-weighted embedding sum
    lla_softmax_wsum<<<NB, 256, 0, stream>>>(all_embs, scores, alpha, lag_f, lag_c, ctxemb);

    // 6) context = ctx_emb @ W_v^T : (64x128, K=512)  [WMMA]  (folds V-projection)
    lla_gemm_at<<<dim3(NB / 16, DV / 16), 32, 0, stream>>>(ctxemb, W_v, out, DM, DV);
}